// ScaledDotProductAttention_21131239097164
// MI455X (gfx1250) — compile-verified
//
#include <hip/hip_runtime.h>

// ---------------------------------------------------------------------------
// Decode attention: q[N,D] x K[N,S,D] -> softmax -> x V[N,S,D]
// N=4096, S=512, D=64, fp32.  HBM-bound (~1.07 GB streamed, ~46us at 23.3TB/s).
// One wave32 per head. Scores via V_WMMA_F32_16X16X4_F32 (fp32-exact, matrix
// unit performs the cross-lane dot-product reduction). p@V via coalesced VALU.
// ---------------------------------------------------------------------------

typedef __attribute__((ext_vector_type(2))) float f32x2;
typedef __attribute__((ext_vector_type(8))) float v8f;

#define NHEADS 4096
#define SEQ    512
#define DIM    64
#define WPB    8          // waves per block
#define NEG_INF (-3.402823466e38f)

__global__ __launch_bounds__(32 * WPB) void
attn_decode_kernel(const float* __restrict__ Q,
                   const float* __restrict__ K,
                   const float* __restrict__ V,
                   float* __restrict__ out,        // [N, D]
                   float* __restrict__ attn_out)   // [N, S]
{
    __shared__ float lds_p[WPB][SEQ];              // scores -> probabilities

    const int lane = threadIdx.x & 31;
    const int wave = threadIdx.x >> 5;
    const int head = blockIdx.x * WPB + wave;

    const float* Kn = K + (size_t)head * SEQ * DIM;
    const float* Vn = V + (size_t)head * SEQ * DIM;
    const float* qn = Q + (size_t)head * DIM;

    // ---- Phase 1: scores[s] = q . K[s,:]  via WMMA f32 16x16x4 ------------
    // A layout (32-bit A 16x4): lanes 0-15 hold rows M=0..15 with K=0,1;
    // lanes 16-31 hold the same rows with K=2,3.  o selects the K-pair.
    const int o = (lane >= 16) ? 2 : 0;
    const int m = lane & 15;

    // B = q broadcast to all 16 columns: per-lane pair {q[4i+o], q[4i+o+1]}
    f32x2 qb[16];
#pragma unroll
    for (int i = 0; i < 16; ++i)
        qb[i] = *(const f32x2*)(qn + 4 * i + o);

    for (int t = 0; t < 32; ++t) {                 // 32 tiles of 16 timesteps
        const int s0 = t * 16;
        const float* Arow = Kn + (size_t)(s0 + m) * DIM + o;
        v8f c = {};
#pragma unroll
        for (int i = 0; i < 16; ++i) {             // D=64 in K-steps of 4
            f32x2 a = __builtin_nontemporal_load((const f32x2*)(Arow + 4 * i));
            c = __builtin_amdgcn_wmma_f32_16x16x4_f32(
                    /*neg_a=*/false, a, /*neg_b=*/false, qb[i],
                    /*c_mod=*/(short)0, c, /*reuse_a=*/false, /*reuse_b=*/false);
        }
        // C/D layout: lane 0 (N=0) holds M=0..7 in v[0..7]; lane 16 holds M=8..15.
        if (m == 0) {
            float* dst = &lds_p[wave][s0 + ((lane >> 4) << 3)];
#pragma unroll
            for (int r = 0; r < 8; ++r) dst[r] = c[r];
        }
    }

    __syncthreads();

    // ---- Phase 2: softmax over S=512 (16 elements per lane) ---------------
    const float scale = 0.125f;                    // 1/sqrt(64)
    float sv[16];
    float mx = NEG_INF;
#pragma unroll
    for (int j = 0; j < 16; ++j) {
        float x = lds_p[wave][lane + 32 * j] * scale;
        sv[j] = x;
        mx = fmaxf(mx, x);
    }
#pragma unroll
    for (int off = 16; off > 0; off >>= 1)
        mx = fmaxf(mx, __shfl_xor(mx, off, 32));

    float sum = 0.0f;
#pragma unroll
    for (int j = 0; j < 16; ++j) {
        float e = __expf(sv[j] - mx);
        sv[j] = e;
        sum += e;
    }
#pragma unroll
    for (int off = 16; off > 0; off >>= 1)
        sum += __shfl_xor(sum, off, 32);
    const float inv = 1.0f / sum;

    float* an = attn_out + (size_t)head * SEQ;
#pragma unroll
    for (int j = 0; j < 16; ++j) {
        float p = sv[j] * inv;
        lds_p[wave][lane + 32 * j] = p;            // keep for phase 3
        an[lane + 32 * j] = p;                     // coalesced attn output
    }

    __syncthreads();

    // ---- Phase 3: out[d] = sum_s p[s] * V[s,d] ----------------------------
    // Lane owns d = {2*lane, 2*lane+1}: each iteration the wave reads one
    // full 256B row of V, perfectly coalesced; p[s] is an LDS broadcast.
    f32x2 acc = {0.0f, 0.0f};
    const float* vp = Vn + lane * 2;
#pragma unroll 4
    for (int s = 0; s < SEQ; ++s) {
        float p  = lds_p[wave][s];
        f32x2 v2 = __builtin_nontemporal_load((const f32x2*)(vp + (size_t)s * DIM));
        acc += v2 * p;
    }
    *(f32x2*)(out + (size_t)head * DIM + lane * 2) = acc;
}

extern "C" void kernel_launch(void* const* d_in, const int* in_sizes, int n_in,
                              void* d_out, int out_size, void* d_ws, size_t ws_size,
                              hipStream_t stream) {
    (void)in_sizes; (void)n_in; (void)out_size; (void)d_ws; (void)ws_size;
    const float* Q = (const float*)d_in[0];
    const float* K = (const float*)d_in[1];
    const float* V = (const float*)d_in[2];
    float* out  = (float*)d_out;                          // first N*D floats
    float* attn = (float*)d_out + (size_t)NHEADS * DIM;   // next  N*S floats

    dim3 grid(NHEADS / WPB);
    dim3 block(32 * WPB);
    attn_decode_kernel<<<grid, block, 0, stream>>>(Q, K, V, out, attn);
}